// CustomSelectAttention_28235115004095
// MI455X (gfx1250) — compile-verified
//
#include <hip/hip_runtime.h>
#include <hip/hip_bf16.h>
#include <math.h>

#define NSLOT 5
#define DK    16
#define DIN   512

typedef __attribute__((ext_vector_type(2))) float v2f;
typedef __attribute__((ext_vector_type(8))) float v8f;

// ---------------- JAX threefry2x32 (default PRNG) ----------------
__device__ __forceinline__ unsigned rotl32(unsigned x, int r) {
    return (x << r) | (x >> (32 - r));
}

__device__ __forceinline__ void threefry2x32(unsigned k0, unsigned k1,
                                             unsigned x0, unsigned x1,
                                             unsigned& o0, unsigned& o1) {
    unsigned ks[3] = {k0, k1, k0 ^ k1 ^ 0x1BD11BDAu};
    x0 += ks[0];
    x1 += ks[1];
    const int rotA[4] = {13, 15, 26, 6};
    const int rotB[4] = {17, 29, 16, 24};
#pragma unroll
    for (int r = 0; r < 5; ++r) {
        const int* rr = (r & 1) ? rotB : rotA;
#pragma unroll
        for (int i = 0; i < 4; ++i) {
            x0 += x1;
            x1 = rotl32(x1, rr[i]);
            x1 ^= x0;
        }
        x0 += ks[(r + 1) % 3];
        x1 += ks[(r + 2) % 3] + (unsigned)(r + 1);
    }
    o0 = x0;
    o1 = x1;
}

// jax.random.uniform(key, (B,5), minval=1e-10, maxval=1.0) element i = b*5+m.
// random_bits for even flat size N splits counts [0..N) into halves:
// bits[i] = out0(i, i+N/2) for i < N/2 else out1(i-N/2, i).
__device__ __forceinline__ float jax_uniform(unsigned key_lo, unsigned i, unsigned half_n) {
    unsigned o0, o1, bits;
    if (i < half_n) {
        threefry2x32(0u, key_lo, i, i + half_n, o0, o1);
        bits = o0;
    } else {
        threefry2x32(0u, key_lo, i - half_n, i, o0, o1);
        bits = o1;
    }
    // bits -> [1,2) -> [0,1): ((bits>>9)|0x3f800000) - 1 ; then u*(1-1e-10)+1e-10
    // (1.0f - 1e-10f) == 1.0f in fp32, so scale reduces to u + 1e-10f.
    float u = __uint_as_float((bits >> 9) | 0x3f800000u) - 1.0f;
    return fmaxf(1e-10f, u + 1e-10f);
}

// gumbel-softmax straight-through hard; writes 5 floats
__device__ __forceinline__ void gumbel_store(const float* s, unsigned key_lo,
                                             unsigned b, unsigned half_n,
                                             float* dst) {
    float y[NSLOT];
#pragma unroll
    for (int m = 0; m < NSLOT; ++m) {
        float u = jax_uniform(key_lo, b * (unsigned)NSLOT + (unsigned)m, half_n);
        float g = -logf(-logf(u));
        y[m] = s[m] + g;  // tau == 1.0
    }
    float mx = y[0];
#pragma unroll
    for (int m = 1; m < NSLOT; ++m) mx = fmaxf(mx, y[m]);
    float e[NSLOT];
    float sum = 0.0f;
#pragma unroll
    for (int m = 0; m < NSLOT; ++m) {
        e[m] = expf(y[m] - mx);
        sum += e[m];
    }
    float inv = 1.0f / sum;
    int idx = 0;
    float best = y[0];
#pragma unroll
    for (int m = 1; m < NSLOT; ++m) {
        if (y[m] > best) { best = y[m]; idx = m; }
    }
#pragma unroll
    for (int m = 0; m < NSLOT; ++m) {
        float soft = e[m] * inv;
        float hard = (m == idx) ? 1.0f : 0.0f;
        dst[m] = (hard - soft) + soft;  // matches stop_gradient(y_hard-y_soft)+y_soft numerics
    }
}

// 224 threads = 7 waves; wave w computes projection slot w (0,1: read slots 0,1;
// 2..6: write slots 0..4) for a tile of 16 batch rows using V_WMMA_F32_16X16X4_F32.
__global__ void __launch_bounds__(224)
select_attn_kernel(const float* __restrict__ q, const float* __restrict__ kk,
                   const float* __restrict__ wr, const float* __restrict__ ww,
                   float* __restrict__ out, int B) {
    __shared__ float sres[7 * 16 * DK];  // 7 slots x 16 rows x 16 cols

    const int tid  = threadIdx.x;
    const int wave = tid >> 5;
    const int lane = tid & 31;
    const int b0   = blockIdx.x * 16;

    // -------- GEMM phase --------
    {
        const int slot = wave;
        const int row  = lane & 15;              // A: M = lane%16
        const int koff = (lane < 16) ? 0 : 2;    // A/B: lane half selects K pair
        const int n    = lane & 15;              // B/C: N = lane%16
        int b = b0 + row;
        if (b >= B) b = B - 1;                   // uniform EXEC for WMMA; dup rows harmless

        const float* arow;
        const float* wmat;
        if (slot < 2) {
            arow = q + ((size_t)b * NSLOT + slot) * DIN;
            wmat = wr + (size_t)slot * DIN * DK;
        } else {
            arow = kk + ((size_t)b * NSLOT + (slot - 2)) * DIN;
            wmat = ww + (size_t)(slot - 2) * DIN * DK;
        }

        v8f c = {};
#pragma unroll 4
        for (int k0 = 0; k0 < DIN; k0 += 4) {
            // A 16x4 f32: lane holds A[M=row][k0+koff], A[M=row][k0+koff+1]
            v2f a = *(const v2f*)(arow + k0 + koff);
            // B 4x16 f32: VGPR0 = B[k0+koff][n], VGPR1 = B[k0+koff+1][n]
            v2f bf;
            bf.x = wmat[(k0 + koff) * DK + n];
            bf.y = wmat[(k0 + koff + 1) * DK + n];
            c = __builtin_amdgcn_wmma_f32_16x16x4_f32(false, a, false, bf,
                                                      (short)0, c, false, false);
        }

        // C layout: VGPR r -> M = r + (lane<16 ? 0 : 8), N = lane%16
        const int mbase = (lane < 16) ? 0 : 8;
        float* dst = &sres[slot * 16 * DK];
#pragma unroll
        for (int r = 0; r < 8; ++r)
            dst[(mbase + r) * DK + n] = c[r];
    }
    __syncthreads();

    // -------- scores + gumbel-softmax-hard phase --------
    if (tid < 16) {
        const int row = tid;
        const int b   = b0 + row;
        if (b < B) {
            const unsigned half_n = (unsigned)((size_t)B * NSLOT / 2);
            float s1[NSLOT], s2[NSLOT];
#pragma unroll
            for (int m = 0; m < NSLOT; ++m) {
                float d1 = 0.0f, d2 = 0.0f;
#pragma unroll
                for (int j = 0; j < DK; ++j) {
                    float wv = sres[(2 + m) * 16 * DK + row * DK + j];
                    d1 += sres[0 * 16 * DK + row * DK + j] * wv;
                    d2 += sres[1 * 16 * DK + row * DK + j] * wv;
                }
                s1[m] = d1 * 0.25f;  // / sqrt(16)
                s2[m] = d2 * 0.25f;
            }
            const size_t halfOut = (size_t)B * NSLOT;
            gumbel_store(s1, 42u, (unsigned)b, half_n, out + (size_t)b * NSLOT);
            gumbel_store(s2, 43u, (unsigned)b, half_n, out + halfOut + (size_t)b * NSLOT);
        }
    }
}

extern "C" void kernel_launch(void* const* d_in, const int* in_sizes, int n_in,
                              void* d_out, int out_size, void* d_ws, size_t ws_size,
                              hipStream_t stream) {
    const float* q  = (const float*)d_in[0];   // [B, 5, 512]
    const float* kk = (const float*)d_in[1];   // [B, 5, 512]
    const float* wr = (const float*)d_in[2];   // [5, 512, 16]
    const float* ww = (const float*)d_in[3];   // [5, 512, 16]
    float* out = (float*)d_out;                // g1 [B,5] then g2 [B,5]

    const int B = in_sizes[0] / (NSLOT * DIN);
    const int grid = (B + 15) / 16;
    select_attn_kernel<<<grid, 224, 0, stream>>>(q, kk, wr, ww, out, B);
}